// FermiNet_18820546691364
// MI455X (gfx1250) — compile-verified
//
#include <hip/hip_runtime.h>
#include <math.h>

#define NE   16
#define NUP  8
#define NA   4
#define EMB  32
#define SGL  256
#define PRD  32
#define NDET 16
#define OIS  322   // one_in LDS row stride (floats): 16 rows conflict-free, 8B aligned
#define HTS  34    // h_two LDS row stride
#define MST  17    // det-matrix LDS row stride
#define BLK  256

#define INV_SQRT2 0.70710678118654752f
#define TANH_GAIN 1.5927812698663606f
#define SILU_GAIN 1.7868129431578026f

typedef float v2f __attribute__((ext_vector_type(2)));
typedef float v8f __attribute__((ext_vector_type(8)));

__device__ __forceinline__ v8f wmma4(v2f a, v2f b, v8f c) {
  // V_WMMA_F32_16X16X4_F32 : D = A(16x4) * B(4x16) + C, full fp32
  return __builtin_amdgcn_wmma_f32_16x16x4_f32(false, a, false, b, (short)0, c,
                                               false, false);
}

__device__ __forceinline__ float silu_gain(float x) {
  return (x / (1.0f + expf(-x))) * SILU_GAIN;
}

struct FNParams {
  const float* electrons;  // (B,16,3)
  const float* atoms;      // (4,3)
  const float* encW;       // (4,32)
  const float* encb;       // (32)
  const float* nucemb;     // (4,32)
  const float* mW0; const float* mb0;   // (32,32),(32)
  const float* mW1; const float* mb1;   // (32,32),(32)
  const float* W1[4];      // (K_l,256)
  const float* b1[4];      // (256)
  const float* Wg[4];      // (2*sdim,256)
  const float* Wp[3];      // (pdim,32)
  const float* bp[3];      // (32)
  const float* Wo_up; const float* sig_up; const float* pi_up;
  const float* Wo_dn; const float* sig_dn; const float* pi_dn;
  const float* wdet;       // (16)
  float* out;              // (B)
};

__global__ __launch_bounds__(BLK) void ferminet_fwd(FNParams P) {
  const int b    = blockIdx.x;
  const int tid  = threadIdx.x;
  const int lane = tid & 31;
  const int wv   = tid >> 5;     // 8 waves
  const int lrow = lane & 15;    // M (A/C) or N (B/C) position
  const int lhi  = lane >> 4;    // half-wave select
  const int koff = lhi << 1;     // K sub-offset for f32 WMMA A/B layout

  __shared__ float s_onein[NE * OIS];      // h_one | g_up | g_dn   (5152 f)
  __shared__ float s_htwo[NE * NE * HTS];  // pair stream           (8704 f)
  __shared__ float s_gterm[SGL];           // glob @ Wg
  __shared__ float s_glob[2 * SGL];
  __shared__ float s_rim4[NE * NA * 4];    // electron-nucleus (dx,dy,dz,|r|)
  __shared__ float s_e[NE * 3];
  __shared__ float s_at[NA * 3];
  __shared__ float s_scratch[4704];        // encoder ping-pong, later det mats
  __shared__ float s_logdet[NDET];
  __shared__ float s_sign[NDET];

  // ---------------- load inputs ----------------
  const float* e = P.electrons + (size_t)b * NE * 3;
  if (tid < NE * 3) s_e[tid] = e[tid];
  if (tid < NA * 3) s_at[tid] = P.atoms[tid];
  __syncthreads();

  // r_im4 (16x4x4)
  if (tid < NE * NA) {
    int i = tid >> 2, m = tid & 3;
    float dx = s_e[i * 3 + 0] - s_at[m * 3 + 0];
    float dy = s_e[i * 3 + 1] - s_at[m * 3 + 1];
    float dz = s_e[i * 3 + 2] - s_at[m * 3 + 2];
    float nr = sqrtf(dx * dx + dy * dy + dz * dz);
    float* o = &s_rim4[(i * NA + m) * 4];
    o[0] = dx; o[1] = dy; o[2] = dz; o[3] = nr;
  }
  // h_two init (16x16 pairs, 4 features), pair row p = i*16 + j
  {
    int i = tid >> 4, j = tid & 15;
    float dx = s_e[i * 3 + 0] - s_e[j * 3 + 0];
    float dy = s_e[i * 3 + 1] - s_e[j * 3 + 1];
    float dz = s_e[i * 3 + 2] - s_e[j * 3 + 2];
    float nr = (i == j) ? 0.0f : sqrtf(dx * dx + dy * dy + dz * dz);
    float* o = &s_htwo[tid * HTS];
    o[0] = dx; o[1] = dy; o[2] = dz; o[3] = nr;
  }
  __syncthreads();

  // ---------------- invariant encoding (VALU, tiny) ----------------
  float* h0 = s_scratch;           // 64 x 33
  float* h1 = s_scratch + 2112;    // 64 x 33
  for (int idx = tid; idx < 64 * EMB; idx += BLK) {
    int rr = idx >> 5, c = idx & 31;
    int i = rr >> 2, m = rr & 3;
    const float* r4 = &s_rim4[(i * NA + m) * 4];
    float acc = P.encb[c] + P.nucemb[m * EMB + c];
#pragma unroll
    for (int k = 0; k < 4; k++) acc += r4[k] * P.encW[k * EMB + c];
    h0[rr * 33 + c] = acc * INV_SQRT2;
  }
  __syncthreads();
  for (int idx = tid; idx < 64 * EMB; idx += BLK) {
    int rr = idx >> 5, c = idx & 31;
    float acc = P.mb0[c];
    for (int k = 0; k < EMB; k++) acc += h0[rr * 33 + k] * P.mW0[k * EMB + c];
    h1[rr * 33 + c] = tanhf(acc) * TANH_GAIN;
  }
  __syncthreads();
  for (int idx = tid; idx < 64 * EMB; idx += BLK) {
    int rr = idx >> 5, c = idx & 31;
    float acc = P.mb1[c];
    for (int k = 0; k < EMB; k++) acc += h1[rr * 33 + k] * P.mW1[k * EMB + c];
    h0[rr * 33 + c] = tanhf(acc) * TANH_GAIN;
  }
  __syncthreads();
  // mean over atoms -> h_one (cols 0..31 of one_in)
  for (int idx = tid; idx < NE * EMB; idx += BLK) {
    int i = idx >> 5, c = idx & 31;
    float acc = 0.0f;
#pragma unroll
    for (int m = 0; m < NA; m++) acc += h0[(i * NA + m) * 33 + c];
    s_onein[i * OIS + c] = acc * 0.25f;
  }
  __syncthreads();

  // ---------------- Fermi layers ----------------
  int sdim = EMB, pdim = 4;
#pragma unroll 1
  for (int l = 0; l < 4; l++) {
    const int K = sdim + 2 * pdim;  // 40 or 320

    // spin-segmented pair means -> one_in cols [sdim, sdim+2*pdim)
    for (int idx = tid; idx < NE * pdim; idx += BLK) {
      int j = idx / pdim, c = idx - j * pdim;
      float su = 0.0f, sd = 0.0f;
#pragma unroll
      for (int i = 0; i < NUP; i++) su += s_htwo[(i * NE + j) * HTS + c];
#pragma unroll
      for (int i = NUP; i < NE; i++) sd += s_htwo[(i * NE + j) * HTS + c];
      s_onein[j * OIS + sdim + c]        = su * 0.125f;
      s_onein[j * OIS + sdim + pdim + c] = sd * 0.125f;
    }
    // glob = [mean(h_one[:8]) | mean(h_one[8:])]
    for (int idx = tid; idx < 2 * sdim; idx += BLK) {
      int c  = (idx < sdim) ? idx : idx - sdim;
      int i0 = (idx < sdim) ? 0 : NUP;
      float acc = 0.0f;
#pragma unroll
      for (int i = 0; i < NUP; i++) acc += s_onein[(i0 + i) * OIS + c];
      s_glob[idx] = acc * 0.125f;
    }
    __syncthreads();

    // gterm = glob @ Wg  (vector x matrix, VALU; column per thread)
    {
      const float* Wg = P.Wg[l];
      float acc = 0.0f;
      for (int k = 0; k < 2 * sdim; k++) acc += s_glob[k] * Wg[k * SGL + tid];
      s_gterm[tid] = acc;
    }

    // ---- one-stream GEMM: (16 x K) @ (K x 256), fp32 WMMA ----
    const float* W1 = P.W1[l];
    v8f acc0 = {}, acc1 = {};
    {
      const int n0 = wv * 32;
      const int c0 = n0 + lrow, c1 = c0 + 16;
      for (int kk = 0; kk < K; kk += 4) {
        v2f a, b0v, b1v;
        int ai = lrow * OIS + kk + koff;
        a.x = s_onein[ai]; a.y = s_onein[ai + 1];
        b0v.x = W1[(kk + koff) * SGL + c0]; b0v.y = W1[(kk + koff + 1) * SGL + c0];
        b1v.x = W1[(kk + koff) * SGL + c1]; b1v.y = W1[(kk + koff + 1) * SGL + c1];
        acc0 = wmma4(a, b0v, acc0);
        acc1 = wmma4(a, b1v, acc1);
      }
    }

    // ---- pair GEMM: (256 x pdim) @ (pdim x 32), fp32 WMMA ----
    v8f pacc[4];
    if (l < 3) {
      const float* Wp = P.Wp[l];
#pragma unroll
      for (int mt = 0; mt < 2; mt++) {
        const int m0 = (wv * 2 + mt) * 16;
        v8f an0 = {}, an1 = {};
        for (int kk = 0; kk < pdim; kk += 4) {
          v2f a, b0v, b1v;
          int ai = (m0 + lrow) * HTS + kk + koff;
          a.x = s_htwo[ai]; a.y = s_htwo[ai + 1];
          b0v.x = Wp[(kk + koff) * PRD + lrow];
          b0v.y = Wp[(kk + koff + 1) * PRD + lrow];
          b1v.x = Wp[(kk + koff) * PRD + 16 + lrow];
          b1v.y = Wp[(kk + koff + 1) * PRD + 16 + lrow];
          an0 = wmma4(a, b0v, an0);
          an1 = wmma4(a, b1v, an1);
        }
        pacc[mt * 2 + 0] = an0;
        pacc[mt * 2 + 1] = an1;
      }
    }
    __syncthreads();  // all reads of old state complete

    // ---- one-stream epilogue: bias + gterm, /sqrt2, silu, residual ----
    {
      const float* b1p = P.b1[l];
#pragma unroll
      for (int t = 0; t < 2; t++) {
        int ncol = wv * 32 + t * 16 + lrow;
        v8f acc = (t == 0) ? acc0 : acc1;
        float bias = b1p[ncol] + s_gterm[ncol];
#pragma unroll
        for (int r = 0; r < 8; r++) {
          int M = r + lhi * 8;
          float pre = (acc[r] + bias) * INV_SQRT2;
          float s = silu_gain(pre);
          float nv = (l == 0) ? s : (s_onein[M * OIS + ncol] + s) * INV_SQRT2;
          s_onein[M * OIS + ncol] = nv;
        }
      }
    }
    // ---- pair epilogue ----
    if (l < 3) {
      const float* bpp = P.bp[l];
#pragma unroll
      for (int mt = 0; mt < 2; mt++) {
#pragma unroll
        for (int nt = 0; nt < 2; nt++) {
          int ncol = nt * 16 + lrow;
          float bias = bpp[ncol];
          v8f acc = pacc[mt * 2 + nt];
#pragma unroll
          for (int r = 0; r < 8; r++) {
            int M = (wv * 2 + mt) * 16 + r + lhi * 8;
            float pre = acc[r] + bias;
            float s = silu_gain(pre);
            float nv = (l == 0) ? s : (s_htwo[M * HTS + ncol] + s) * INV_SQRT2;
            s_htwo[M * HTS + ncol] = nv;
          }
        }
      }
    }
    __syncthreads();
    sdim = SGL; pdim = PRD;
  }

  // ---------------- orbitals + envelope -> det matrices ----------------
  float* mats = s_scratch;  // 16 dets x (16 x 17)
#pragma unroll 1
  for (int spin = 0; spin < 2; spin++) {
    const float* Wo  = spin ? P.Wo_dn  : P.Wo_up;
    const float* sig = spin ? P.sig_dn : P.sig_up;
    const float* pi  = spin ? P.pi_dn  : P.pi_up;
    const int o0 = wv * 32 + lrow, o1 = o0 + 16;

    v8f acc0 = {}, acc1 = {};
    for (int kk = 0; kk < SGL; kk += 4) {
      v2f a, b0v, b1v;
      int ai = lrow * OIS + kk + koff;
      a.x = s_onein[ai]; a.y = s_onein[ai + 1];
      b0v.x = Wo[(kk + koff) * SGL + o0]; b0v.y = Wo[(kk + koff + 1) * SGL + o0];
      b1v.x = Wo[(kk + koff) * SGL + o1]; b1v.y = Wo[(kk + koff + 1) * SGL + o1];
      acc0 = wmma4(a, b0v, acc0);
      acc1 = wmma4(a, b1v, acc1);
    }

    bool valid_half = (spin == 0) ? (lhi == 0) : (lhi == 1);
#pragma unroll
    for (int t = 0; t < 2; t++) {
      int o = (t == 0) ? o0 : o1;
      v8f acc = (t == 0) ? acc0 : acc1;
      if (valid_half) {
        // hoist sigma / pi loads (independent of electron row)
        float sgv[9 * NA], piv[NA];
#pragma unroll
        for (int kp = 0; kp < 9; kp++)
#pragma unroll
          for (int m = 0; m < NA; m++)
            sgv[kp * NA + m] = sig[(kp * NA + m) * SGL + o];
#pragma unroll
        for (int m = 0; m < NA; m++) piv[m] = pi[m * SGL + o];

        int d = o >> 4, jc = o & 15;
#pragma unroll
        for (int r = 0; r < 8; r++) {
          int i = r + lhi * 8;  // electron index
          float env = 0.0f;
#pragma unroll
          for (int m = 0; m < NA; m++) {
            const float* r4 = &s_rim4[(i * NA + m) * 4];
            float v0 = 0.0f, v1 = 0.0f, v2 = 0.0f;
#pragma unroll
            for (int k = 0; k < 3; k++) {
              float rv = r4[k];
              v0 += rv * sgv[(k * 3 + 0) * NA + m];
              v1 += rv * sgv[(k * 3 + 1) * NA + m];
              v2 += rv * sgv[(k * 3 + 2) * NA + m];
            }
            env += expf(-sqrtf(v0 * v0 + v1 * v1 + v2 * v2)) * piv[m];
          }
          mats[(d * 16 + i) * MST + jc] = acc[r] * env;
        }
      }
    }
  }
  __syncthreads();

  // ---------------- slogdet: per wave, 2 dets (one per half-wave) ----------
  {
    const int d = wv * 2 + lhi;
    const int j = lrow;  // this lane owns column j of det d
    float* A = &mats[d * 16 * MST];
    float sgn = 1.0f, ld = 0.0f;
    for (int k = 0; k < 16; k++) {
      // partial pivot (redundant per half-wave, broadcast LDS reads)
      int p = k;
      float best = fabsf(A[k * MST + k]);
      for (int i = k + 1; i < 16; i++) {
        float v = fabsf(A[i * MST + k]);
        if (v > best) { best = v; p = i; }
      }
      if (p != k) {
        float tmp = A[k * MST + j];
        A[k * MST + j] = A[p * MST + j];
        A[p * MST + j] = tmp;
        sgn = -sgn;
      }
      __builtin_amdgcn_wave_barrier();
      float akk = A[k * MST + k];
      float akj = A[k * MST + j];
      ld += logf(fabsf(akk));
      if (akk < 0.0f) sgn = -sgn;
      float rakk = 1.0f / akk;
      for (int i = k + 1; i < 16; i++) {
        float mlt = A[i * MST + k] * rakk;  // read col k (pre-update this step)
        if (j > k) A[i * MST + j] -= mlt * akj;
      }
      __builtin_amdgcn_wave_barrier();
    }
    if (j == 0) { s_logdet[d] = ld; s_sign[d] = sgn; }
  }
  __syncthreads();

  // ---------------- log-sum-det combine ----------------
  if (tid == 0) {
    float mx = s_logdet[0];
#pragma unroll
    for (int d = 1; d < NDET; d++) mx = fmaxf(mx, s_logdet[d]);
    float res = 0.0f;
#pragma unroll
    for (int d = 0; d < NDET; d++)
      res += P.wdet[d] * s_sign[d] * expf(s_logdet[d] - mx);
    P.out[b] = logf(fabsf(res)) + mx;
  }
}

extern "C" void kernel_launch(void* const* d_in, const int* in_sizes, int n_in,
                              void* d_out, int out_size, void* d_ws, size_t ws_size,
                              hipStream_t stream) {
  (void)n_in; (void)d_ws; (void)ws_size; (void)out_size;
  const float* const* f = (const float* const*)d_in;
  FNParams P;
  P.electrons = f[0];
  P.atoms     = f[1];
  P.encW      = f[2];
  P.encb      = f[3];
  P.nucemb    = f[4];
  P.mW0 = f[5]; P.mb0 = f[6];
  P.mW1 = f[7]; P.mb1 = f[8];
  int idx = 9;
  for (int l = 0; l < 4; l++) {
    P.W1[l] = f[idx++];
    P.b1[l] = f[idx++];
    P.Wg[l] = f[idx++];
    if (l < 3) { P.Wp[l] = f[idx++]; P.bp[l] = f[idx++]; }
  }
  P.Wo_up = f[idx++]; P.sig_up = f[idx++]; P.pi_up = f[idx++];
  P.Wo_dn = f[idx++]; P.sig_dn = f[idx++]; P.pi_dn = f[idx++];
  P.wdet  = f[idx++];
  P.out   = (float*)d_out;

  const int B = in_sizes[0] / (NE * 3);  // 2048
  ferminet_fwd<<<B, BLK, 0, stream>>>(P);
}